// MixLoraRouter_59717225284240
// MI455X (gfx1250) — compile-verified
//
#include <hip/hip_runtime.h>
#include <hip/hip_bf16.h>

typedef __attribute__((ext_vector_type(2))) float v2f;
typedef __attribute__((ext_vector_type(8))) float v8f;

#define NUM_EXPERTS 8
#define HIDDEN 4096
#define NUM_TOKENS 16384
#define TOKENS_PER_BLOCK 128
#define THREADS 256

// Main kernel: one wave computes a 16-token x 16-expert logit tile via
// V_WMMA_F32_16X16X4_F32 over K=4096, then does softmax/top-2 per token.
__global__ __launch_bounds__(THREADS) void moe_router_kernel(
    const float* __restrict__ H,   // [16384, 4096]
    const float* __restrict__ G,   // [8, 4096]
    float* __restrict__ out,       // [weights 32768][indices 32768][aux 1]
    float* __restrict__ ws)        // [cnt 8][psum 8]
{
    extern __shared__ float lds[];
    float* lds_gate   = lds;                          // 8*4096 floats (128 KB)
    float* lds_logits = lds + NUM_EXPERTS * HIDDEN;   // 8 waves * 16 tok * 8 exp
    float* lds_cnt    = lds_logits + 8 * 16 * 8;      // 8
    float* lds_psum   = lds_cnt + 8;                  // 8

    const int tid = threadIdx.x;

    // Cooperative copy of the gate matrix into LDS (it is reused by every
    // K-step of every wave; 128 KB fits easily in the 320 KB WGP LDS).
    {
        const float4* g4 = (const float4*)G;
        float4* l4 = (float4*)lds_gate;
        const int n4 = NUM_EXPERTS * HIDDEN / 4;
        for (int i = tid; i < n4; i += THREADS) l4[i] = g4[i];
    }
    if (tid < 8) { lds_cnt[tid] = 0.0f; lds_psum[tid] = 0.0f; }
    __syncthreads();

    const int lane  = tid & 31;
    const int wave  = tid >> 5;
    const int half  = lane >> 4;     // 0: lanes 0-15, 1: lanes 16-31
    const int n     = lane & 15;     // A: row M; B/C: column N (expert)
    const int token0 = blockIdx.x * TOKENS_PER_BLOCK + wave * 16;

    // A fragment (32-bit A 16x4): lanes 0-15 carry K={k,k+1} of row M=n,
    // lanes 16-31 carry K={k+2,k+3}.  One contiguous 8B load per step.
    const float* aRow = H + (size_t)(token0 + n) * HIDDEN + half * 2;
    // B fragment (4x16): lanes = N, lower half rows K={k,k+1}, upper K={k+2,k+3}.
    // B[k][e] = G[e][k]; experts 8..15 are masked to zero (keeps EXEC all-1s).
    const float* bRow = lds_gate + (size_t)(n & 7) * HIDDEN + half * 2;
    const float bmask = (n < NUM_EXPERTS) ? 1.0f : 0.0f;

    v8f acc = {0.f, 0.f, 0.f, 0.f, 0.f, 0.f, 0.f, 0.f};

    for (int kb = 0; kb < HIDDEN; kb += 64) {
        // Hint the next 256B of the activation stream toward the caches.
        __builtin_prefetch(aRow + kb + 64, 0, 1);
#pragma unroll
        for (int kk = 0; kk < 64; kk += 4) {
            v2f a = *(const v2f*)(aRow + kb + kk);   // global_load_b64
            v2f b = *(const v2f*)(bRow + kb + kk);   // ds_load_b64
            b *= bmask;
            acc = __builtin_amdgcn_wmma_f32_16x16x4_f32(
                false, a, false, b, (short)0, acc, false, false);
        }
    }

    // C/D layout: VGPR r holds M=r (lanes 0-15) / M=r+8 (lanes 16-31), N=lane&15.
    // Spill the 8 valid expert columns to LDS so one lane can own one token.
    if (n < NUM_EXPERTS) {
        float* dst = lds_logits + wave * 128 + half * 64 + n;  // [tok_local][exp]
#pragma unroll
        for (int r = 0; r < 8; ++r) dst[r * 8] = acc[r];
    }
    __syncthreads();

    if (lane < 16) {
        const int token = token0 + lane;
        const float* lg = lds_logits + wave * 128 + lane * 8;

        float l[NUM_EXPERTS], m = lg[0];
#pragma unroll
        for (int e = 0; e < NUM_EXPERTS; ++e) { l[e] = lg[e]; m = fmaxf(m, l[e]); }
        float p[NUM_EXPERTS], s = 0.0f;
#pragma unroll
        for (int e = 0; e < NUM_EXPERTS; ++e) { p[e] = __expf(l[e] - m); s += p[e]; }
        const float inv = 1.0f / s;
#pragma unroll
        for (int e = 0; e < NUM_EXPERTS; ++e) p[e] *= inv;

        // top-2 (stable: first index wins ties, matching jax.lax.top_k)
        int i1 = 0; float v1 = p[0];
#pragma unroll
        for (int e = 1; e < NUM_EXPERTS; ++e) if (p[e] > v1) { v1 = p[e]; i1 = e; }
        int i2 = -1; float v2 = -1.0f;
#pragma unroll
        for (int e = 0; e < NUM_EXPERTS; ++e)
            if (e != i1 && p[e] > v2) { v2 = p[e]; i2 = e; }

        const float rs = 1.0f / (v1 + v2);
        out[token * 2 + 0] = v1 * rs;
        out[token * 2 + 1] = v2 * rs;
        out[NUM_TOKENS * 2 + token * 2 + 0] = (float)i1;
        out[NUM_TOKENS * 2 + token * 2 + 1] = (float)i2;

        atomicAdd(&lds_cnt[i1], 1.0f);
        atomicAdd(&lds_cnt[i2], 1.0f);
#pragma unroll
        for (int e = 0; e < NUM_EXPERTS; ++e) atomicAdd(&lds_psum[e], p[e]);
    }
    __syncthreads();

    if (tid < 8) {
        atomicAdd(&ws[tid], lds_cnt[tid]);
        atomicAdd(&ws[8 + tid], lds_psum[tid]);
    }
}

__global__ void moe_router_finalize(const float* __restrict__ ws,
                                    float* __restrict__ out)
{
    if (threadIdx.x == 0 && blockIdx.x == 0) {
        const float invT = 1.0f / (float)NUM_TOKENS;
        float s = 0.0f;
#pragma unroll
        for (int e = 0; e < NUM_EXPERTS; ++e)
            s += (ws[e] * invT) * (ws[8 + e] * invT);
        out[NUM_TOKENS * 4] = (float)NUM_EXPERTS * s;  // aux_loss
    }
}

extern "C" void kernel_launch(void* const* d_in, const int* in_sizes, int n_in,
                              void* d_out, int out_size, void* d_ws, size_t ws_size,
                              hipStream_t stream) {
    const float* H = (const float*)d_in[0];   // hidden_states [16384,4096] f32
    const float* G = (const float*)d_in[1];   // gate_weight   [8,4096]     f32
    float* out = (float*)d_out;
    float* ws  = (float*)d_ws;

    // zero the 16-float accumulator workspace (graph-capture safe)
    hipMemsetAsync(ws, 0, 16 * sizeof(float), stream);

    const size_t lds_bytes =
        (NUM_EXPERTS * HIDDEN + 8 * 16 * 8 + 16) * sizeof(float);  // ~135 KB
    dim3 grid(NUM_TOKENS / TOKENS_PER_BLOCK);  // 128
    dim3 block(THREADS);                       // 256 = 8 waves
    moe_router_kernel<<<grid, block, lds_bytes, stream>>>(H, G, out, ws);
    moe_router_finalize<<<1, 32, 0, stream>>>(ws, out);
}